// StagVI_NodeClassification_R1_23021024707491
// MI455X (gfx1250) — compile-verified
//
#include <hip/hip_runtime.h>
#include <hip/hip_bf16.h>

typedef __attribute__((ext_vector_type(2))) float v2f;
typedef __attribute__((ext_vector_type(8))) float v8f;

#define N_NODES 50000
#define N_EDGES 800000
#define FEAT    64
#define OUTF    16
#define BN_EPS  1e-5f

// ---------------------------------------------------------------- zero fill
__global__ void zero_kernel(float* __restrict__ p, int n) {
    int i = blockIdx.x * blockDim.x + threadIdx.x;
    int stride = gridDim.x * blockDim.x;
    for (; i < n; i += stride) p[i] = 0.0f;
}

// ------------------------------------------------- per-edge reweight+scatter
// m[e,f] = xin[src[e],f] * (a_mu + sigma*eps[e,f]);  agg[dst[e],f] += m
// one thread per (edge, 4-feature group), float4 vectorized
__global__ void edge_scatter_kernel(const float* __restrict__ xin,
                                    const float* __restrict__ eps,
                                    const int*   __restrict__ src,
                                    const int*   __restrict__ dst,
                                    const float* __restrict__ a_mu,
                                    const float* __restrict__ a_log_sigma,
                                    float* __restrict__ agg) {
    const int tid = blockIdx.x * blockDim.x + threadIdx.x;
    const int total = N_EDGES * (FEAT / 4);
    if (tid >= total) return;
    const int e  = tid >> 4;          // FEAT/4 == 16 groups per edge
    const int fg = (tid & 15) << 2;   // feature offset 0,4,...,60

    const float mu = a_mu[0];
    const float sg = __expf(a_log_sigma[0]);

    const int s = src[e];
    const int d = dst[e];

    const float4 xe = *(const float4*)(xin + (size_t)s * FEAT + fg);
    const float4 ep = *(const float4*)(eps + (size_t)e * FEAT + fg);

    float4 m;
    m.x = xe.x * (mu + sg * ep.x);
    m.y = xe.y * (mu + sg * ep.y);
    m.z = xe.z * (mu + sg * ep.z);
    m.w = xe.w * (mu + sg * ep.w);

    float* base = agg + (size_t)d * FEAT + fg;
    atomicAdd(base + 0, m.x);
    atomicAdd(base + 1, m.y);
    atomicAdd(base + 2, m.z);
    atomicAdd(base + 3, m.w);
}

// ---------------------------------------------------- fp32 WMMA tiled GEMM
// out[M x Nc] = act(A[M x K] @ W[K x Nc] + bias), one 16x16 tile per wave,
// K stepped by 4 via V_WMMA_F32_16X16X4_F32 (exact fp32 accumulate).
__global__ void gemm_wmma_f32_kernel(const float* __restrict__ A,
                                     const float* __restrict__ W,
                                     const float* __restrict__ bias,
                                     float* __restrict__ out,
                                     int M, int K, int Nc,
                                     int relu, int ntiles) {
    const int wave = threadIdx.x >> 5;             // 4 waves per block
    const int tile = blockIdx.x * 4 + wave;
    if (tile >= ntiles) return;

    const int colTiles = Nc >> 4;
    const int tm = tile / colTiles;
    const int tn = tile % colTiles;

    const int lane = threadIdx.x & 31;
    const int half = lane >> 4;        // 0: lanes 0-15, 1: lanes 16-31
    const int l16  = lane & 15;

    const int arow = tm * 16 + l16;    // this lane's A row
    const int bcol = tn * 16 + l16;    // this lane's B/C column

    v8f acc = {};
    for (int k = 0; k < K; k += 4) {
        const int ka = k + half * 2;   // lane-half K split per ISA layout
        v2f a, b;
        a.x = A[(size_t)arow * K + ka];
        a.y = A[(size_t)arow * K + ka + 1];
        b.x = W[(size_t)(ka)     * Nc + bcol];
        b.y = W[(size_t)(ka + 1) * Nc + bcol];
        acc = __builtin_amdgcn_wmma_f32_16x16x4_f32(
            /*neg_a=*/false, a, /*neg_b=*/false, b,
            /*c_mod=*/(short)0, acc, /*reuse_a=*/false, /*reuse_b=*/false);
    }

    const float bs = bias[bcol];
#pragma unroll
    for (int r = 0; r < 8; ++r) {
        const int orow = tm * 16 + r + half * 8;   // C/D VGPR layout
        float v = acc[r] + bs;
        if (relu) v = fmaxf(v, 0.0f);
        out[(size_t)orow * Nc + bcol] = v;
    }
}

// ------------------------------------------------------- batchnorm statistics
// stats[0..63] = sum over rows, stats[64..127] = sum of squares
__global__ void bn_stats_kernel(const float* __restrict__ t,
                                float* __restrict__ stats, int M) {
    const int f   = threadIdx.x & 63;
    const int sub = threadIdx.x >> 6;   // 0..3 (block of 256)
    float s = 0.0f, s2 = 0.0f;
    for (int r = blockIdx.x * 4 + sub; r < M; r += gridDim.x * 4) {
        float v = t[(size_t)r * FEAT + f];
        s  += v;
        s2 += v * v;
    }
    __shared__ float ls[256], ls2[256];
    ls[threadIdx.x]  = s;
    ls2[threadIdx.x] = s2;
    __syncthreads();
    if (sub == 0) {
        s  = ls[f]  + ls[64 + f]  + ls[128 + f]  + ls[192 + f];
        s2 = ls2[f] + ls2[64 + f] + ls2[128 + f] + ls2[192 + f];
        atomicAdd(&stats[f],      s);
        atomicAdd(&stats[64 + f], s2);
    }
}

// ------------------------------------------------- batchnorm + relu, in place
__global__ void bn_relu_kernel(float* __restrict__ t,
                               const float* __restrict__ stats,
                               const float* __restrict__ gamma,
                               const float* __restrict__ beta, int M) {
    int i = blockIdx.x * blockDim.x + threadIdx.x;
    const int n = M * FEAT;
    if (i >= n) return;
    const int f = i & 63;
    const float invM = 1.0f / (float)M;
    const float mean = stats[f] * invM;
    const float var  = stats[64 + f] * invM - mean * mean;
    const float inv  = rsqrtf(var + BN_EPS);
    const float v    = gamma[f] * (t[i] - mean) * inv + beta[f];
    t[i] = fmaxf(v, 0.0f);
}

// ---------------------------------------------------------------------------
extern "C" void kernel_launch(void* const* d_in, const int* in_sizes, int n_in,
                              void* d_out, int out_size, void* d_ws, size_t ws_size,
                              hipStream_t stream) {
    const float* x     = (const float*)d_in[0];
    const float* eps0  = (const float*)d_in[1];
    const float* eps1  = (const float*)d_in[2];
    const float* eps2  = (const float*)d_in[3];
    const float* W0    = (const float*)d_in[4];
    const float* b0    = (const float*)d_in[5];
    const float* W1    = (const float*)d_in[6];
    const float* b1    = (const float*)d_in[7];
    const float* W2    = (const float*)d_in[8];
    const float* b2    = (const float*)d_in[9];
    const float* gamma1= (const float*)d_in[10];
    const float* beta1 = (const float*)d_in[11];
    const float* a_mu  = (const float*)d_in[12];
    const float* a_ls  = (const float*)d_in[13];
    const int*   src   = (const int*)d_in[14];
    const int*   dst   = (const int*)d_in[15];

    const int nodeElems = N_NODES * FEAT;          // 3.2M floats = 12.8 MB
    float* agg   = (float*)d_ws;
    float* hbuf  = agg  + nodeElems;
    float* tbuf  = hbuf + nodeElems;
    float* stats = tbuf + nodeElems;               // 128 floats

    const int edgeThreads = N_EDGES * (FEAT / 4);  // 12.8M
    const dim3 edgeGrid((edgeThreads + 255) / 256);
    const dim3 zeroGrid(2048);

    const int tiles64 = (N_NODES / 16) * (FEAT / 16);   // 3125*4 = 12500
    const int tiles16 = (N_NODES / 16) * (OUTF / 16);   // 3125
    const dim3 gemmGrid64((tiles64 + 3) / 4);
    const dim3 gemmGrid16((tiles16 + 3) / 4);

    // ---- layer 0: conv + relu -------------------------------------------
    zero_kernel<<<zeroGrid, 256, 0, stream>>>(agg, nodeElems);
    edge_scatter_kernel<<<edgeGrid, 256, 0, stream>>>(x, eps0, src, dst, a_mu, a_ls, agg);
    gemm_wmma_f32_kernel<<<gemmGrid64, 128, 0, stream>>>(agg, W0, b0, hbuf,
                                                         N_NODES, FEAT, FEAT, 1, tiles64);

    // ---- layer 1: conv + batchnorm + relu -------------------------------
    zero_kernel<<<zeroGrid, 256, 0, stream>>>(agg, nodeElems);
    edge_scatter_kernel<<<edgeGrid, 256, 0, stream>>>(hbuf, eps1, src, dst, a_mu, a_ls, agg);
    gemm_wmma_f32_kernel<<<gemmGrid64, 128, 0, stream>>>(agg, W1, b1, tbuf,
                                                         N_NODES, FEAT, FEAT, 0, tiles64);
    zero_kernel<<<dim3(1), 128, 0, stream>>>(stats, 128);
    bn_stats_kernel<<<dim3(256), 256, 0, stream>>>(tbuf, stats, N_NODES);
    bn_relu_kernel<<<dim3((nodeElems + 255) / 256), 256, 0, stream>>>(tbuf, stats,
                                                                      gamma1, beta1, N_NODES);

    // ---- layer 2: conv -> logits ----------------------------------------
    zero_kernel<<<zeroGrid, 256, 0, stream>>>(agg, nodeElems);
    edge_scatter_kernel<<<edgeGrid, 256, 0, stream>>>(tbuf, eps2, src, dst, a_mu, a_ls, agg);
    gemm_wmma_f32_kernel<<<gemmGrid16, 128, 0, stream>>>(agg, W2, b2, (float*)d_out,
                                                         N_NODES, FEAT, OUTF, 0, tiles16);
}